// BoundaryLoss2_19696720019759
// MI455X (gfx1250) — compile-verified
//
#include <hip/hip_runtime.h>
#include <math.h>

#define BB   8
#define HH   256
#define WW   256
#define NPIX (BB * HH * WW)   // 524288
#define NBLK (BB * HH)        // 2048

typedef float v2f __attribute__((ext_vector_type(2)));
typedef float v8f __attribute__((ext_vector_type(8)));

// Deterministic wave32 reduction of n floats (n multiple of 64) at src[i*stride]
// using V_WMMA_F32_16X16X4_F32 with B = ones:
//   D[m,n] = sum_k A[m,k] + C[m,n]  -> every D column holds 16 row-sums.
// Lane n (<16) holds D[0..7][n], lane n+16 holds D[8..15][n]; one xor-16
// shuffle completes the total on every lane. Exact f32 arithmetic, fixed order.
__device__ __forceinline__ float wave_wmma_sum(const float* src, int n, int stride) {
  int lane = (int)(threadIdx.x & 31);
  v2f bmat;
  bmat.x = 1.0f;
  bmat.y = 1.0f;
  v8f c = {};
  for (int base = 0; base < n; base += 64) {
    v2f a;
    a.x = src[(size_t)(base + lane) * (size_t)stride];
    a.y = src[(size_t)(base + 32 + lane) * (size_t)stride];
    c = __builtin_amdgcn_wmma_f32_16x16x4_f32(false, a, false, bmat,
                                              (short)0, c, false, false);
  }
  float p = c[0] + c[1] + c[2] + c[3] + c[4] + c[5] + c[6] + c[7];
  p += __shfl_xor(p, 16, 32);
  return p;
}

// Kernel 1: vertical nearest-True scan (exact replica of reference scan),
// for both masks (s=0: t==1 "pos", s=1: t==0 "neg"). Writes g^2.
// One thread per (mask, batch, column); lanes cover adjacent columns -> coalesced.
__global__ void vert_scan_kernel(const float* __restrict__ targets,
                                 float* __restrict__ g2) {
  int tid = blockIdx.x * blockDim.x + threadIdx.x;  // 0 .. 2*BB*WW-1
  int j = tid & (WW - 1);
  int b = (tid >> 8) & (BB - 1);
  int s = tid >> 11;

  const float* tcol = targets + (size_t)b * HH * WW + j;
  float* gcol = g2 + (size_t)s * NPIX + (size_t)b * HH * WW + j;

  const float BIG = 65536.0f;  // H*W, as in reference
  float carry = BIG;
  for (int r = 0; r < HH; ++r) {
    float t = tcol[(size_t)r * WW];
    bool m = (s == 0) ? (t > 0.5f) : !(t > 0.5f);
    float d = m ? 0.0f : (carry + 1.0f);
    gcol[(size_t)r * WW] = d;
    carry = d;
  }
  carry = BIG;
  for (int r = HH - 1; r >= 0; --r) {
    float t = tcol[(size_t)r * WW];
    bool m = (s == 0) ? (t > 0.5f) : !(t > 0.5f);
    float d = m ? 0.0f : (carry + 1.0f);
    carry = d;
    float g = fminf(d, gcol[(size_t)r * WW]);
    gcol[(size_t)r * WW] = g * g;
  }
}

// Kernel 2: one block per image row. LDS-resident min-plus EDT column pass,
// fused sigmoid / BCE / dice / boundary terms, WMMA block reduction.
__global__ void __launch_bounds__(256) row_loss_kernel(
    const float* __restrict__ logits, const float* __restrict__ targets,
    const float* __restrict__ g2, float* __restrict__ partials) {
  int bh = blockIdx.x;       // b*HH + h
  int j = threadIdx.x;       // column
  __shared__ float g2i[2 * WW];   // interleaved pos/neg -> conflict-free selects
  __shared__ float red[5][WW];

  size_t rowoff = (size_t)bh * WW;
  g2i[2 * j + 0] = g2[rowoff + j];                 // g^2 of pos mask
  g2i[2 * j + 1] = g2[(size_t)NPIX + rowoff + j];  // g^2 of neg mask
  __syncthreads();

  float t = targets[rowoff + j];
  float x = logits[rowoff + j];
  float sig = 1.0f / (1.0f + expf(-x));
  // softplus(x) - x*t, overflow-safe like jax.nn.softplus
  float bce = fmaxf(x, 0.0f) + log1pf(expf(-fabsf(x))) - x * t;

  // pos pixel needs distance-to-neg; neg pixel needs distance-to-pos
  int sel = (t > 0.5f) ? 1 : 0;
  float jf = (float)j;
  float d2 = 3.402823466e38f;
#pragma unroll 8
  for (int k = 0; k < WW; ++k) {
    float diff = jf - (float)k;
    d2 = fminf(d2, g2i[2 * k + sel] + diff * diff);
  }
  float dv = sqrtf(d2);
  float dist = (t > 0.5f) ? (1.0f - dv) : dv;  // res = dto*neg - (dtn-1)*pos

  red[0][j] = sig * t;     // dice intersection
  red[1][j] = sig;         // sum(sig)
  red[2][j] = t;           // sum(t) (also has_pos source)
  red[3][j] = bce;         // bce sum
  red[4][j] = sig * dist;  // boundary sum (gated later by has_pos)
  __syncthreads();

  if (j < 32) {  // wave 0, EXEC all-ones for WMMA
    for (int c = 0; c < 5; ++c) {
      float s = wave_wmma_sum(&red[c][0], WW, 1);
      if (j == 0) partials[(size_t)bh * 8 + c] = s;
    }
  }
}

// Kernel 3: single wave. WMMA-sum the 2048 block partials; per-batch
// has_pos gating for the boundary term; emit the scalar loss.
__global__ void finalize_kernel(const float* __restrict__ partials,
                                float* __restrict__ out) {
  float inter = wave_wmma_sum(partials + 0, NBLK, 8);
  float ssig  = wave_wmma_sum(partials + 1, NBLK, 8);
  float sbce  = wave_wmma_sum(partials + 3, NBLK, 8);

  float stt = 0.0f, sbdy = 0.0f;
  for (int b = 0; b < BB; ++b) {
    const float* pb = partials + (size_t)b * HH * 8;
    float tb = wave_wmma_sum(pb + 2, HH, 8);  // exact: integer-valued sums
    float db = wave_wmma_sum(pb + 4, HH, 8);
    stt += tb;
    sbdy += (tb > 0.0f) ? db : 0.0f;  // has_pos per batch
  }

  if (threadIdx.x == 0) {
    float uni = ssig + stt + 1e-5f;
    float dice = 1.0f - (2.0f * inter + 1e-5f) / uni;
    float bce = sbce * (1.0f / (float)NPIX);
    float bdy = sbdy * (1.0f / (float)NPIX);
    out[0] = 0.5f * dice + 0.5f * bce + 0.5f * bdy;
  }
}

extern "C" void kernel_launch(void* const* d_in, const int* in_sizes, int n_in,
                              void* d_out, int out_size, void* d_ws, size_t ws_size,
                              hipStream_t stream) {
  (void)in_sizes; (void)n_in; (void)out_size; (void)ws_size;
  const float* logits  = (const float*)d_in[0];
  const float* targets = (const float*)d_in[1];

  float* g2 = (float*)d_ws;                       // 2 * NPIX floats (4 MB)
  float* partials = g2 + 2 * (size_t)NPIX;        // NBLK * 8 floats (64 KB)

  vert_scan_kernel<<<(2 * BB * WW) / 256, 256, 0, stream>>>(targets, g2);
  row_loss_kernel<<<NBLK, 256, 0, stream>>>(logits, targets, g2, partials);
  finalize_kernel<<<1, 32, 0, stream>>>(partials, (float*)d_out);
}